// AdjacencyMatrix_58660663329279
// MI455X (gfx1250) — compile-verified
//
#include <hip/hip_runtime.h>
#include <hip/hip_bf16.h>
#include <math.h>

#define V 8
#define B 32
#define CIN 8
#define T 4096
#define O 64
#define H 16

typedef float v2f __attribute__((ext_vector_type(2)));
typedef float v8f __attribute__((ext_vector_type(8)));

__device__ __forceinline__ float selu_f(float x) {
  const float scale = 1.0507009873554805f;
  const float alpha = 1.6732632423543772f;
  return x > 0.0f ? scale * x : scale * alpha * (expf(x) - 1.0f);
}

// ---------------------------------------------------------------------------
// Kernel A: one block per (v,b). Reduce x over C_IN into xs[v,b,:] (kept in
// LDS + written to workspace), then compute s[v,b,o] = mean_t selu(conv).
// ---------------------------------------------------------------------------
__global__ void conv_stats_kernel(const float* __restrict__ x,
                                  const float* __restrict__ conv_w,
                                  const float* __restrict__ conv_b,
                                  float* __restrict__ xs,
                                  float* __restrict__ s) {
  __shared__ float xrow[T];
  const int vb = blockIdx.x;
  const int tid = threadIdx.x;
  const float* xp = x + (size_t)vb * CIN * T;
  for (int t = tid; t < T; t += 256) {
    float sum = 0.f;
#pragma unroll
    for (int c = 0; c < CIN; ++c) sum += xp[c * T + t];
    xrow[t] = sum;
    xs[(size_t)vb * T + t] = sum;
  }
  __syncthreads();
  // 256 threads = 64 o's x 4 t-quarters
  const int o = tid >> 2;
  const int q = tid & 3;
  const float w0 = conv_w[o * 5 + 0], w1 = conv_w[o * 5 + 1],
              w2 = conv_w[o * 5 + 2], w3 = conv_w[o * 5 + 3],
              w4 = conv_w[o * 5 + 4];
  const float bo = conv_b[o];
  const int t0 = q * (T / 4);
  // rolling window over xrow[t-2 .. t+2] with zero padding
  float a0 = (t0 >= 2) ? xrow[t0 - 2] : 0.f;
  float a1 = (t0 >= 1) ? xrow[t0 - 1] : 0.f;
  float a2 = xrow[t0];
  float a3 = xrow[t0 + 1];
  float a4 = xrow[t0 + 2];
  float acc = 0.f;
  for (int t = t0; t < t0 + T / 4; ++t) {
    float cv = fmaf(a0, w0, fmaf(a1, w1, fmaf(a2, w2,
               fmaf(a3, w3, fmaf(a4, w4, bo)))));
    acc += selu_f(cv);
    a0 = a1; a1 = a2; a2 = a3; a3 = a4;
    const int tn = t + 3;
    a4 = (tn < T) ? xrow[tn] : 0.f;
  }
  // reduce the 4 quarters (lanes tid^1, tid^2 are in the same wave32)
  acc += __shfl_xor(acc, 1);
  acc += __shfl_xor(acc, 2);
  if (q == 0) s[(size_t)vb * O + o] = acc * (1.0f / T);
}

// ---------------------------------------------------------------------------
// Kernel B: single block, 8 waves (wave w handles vertex v=w).
// WMMA f32 16x16x4 for both SE matmuls; gate/compressed/attention-dot fused
// into the second matmul epilogue; softmax over v_from at the end.
// ---------------------------------------------------------------------------
__global__ void se_attn_kernel(const float* __restrict__ s,
                               const float* __restrict__ se_w1,
                               const float* __restrict__ se_b1,
                               const float* __restrict__ se_w2,
                               const float* __restrict__ se_b2,
                               const float* __restrict__ attn_w,
                               float* __restrict__ sm) {
  __shared__ float hbuf[V][B][H + 1];  // relu(s@w1^T + b1), padded
  __shared__ float uu[V][B];           // dot(compressed, attn_w[:O])
  __shared__ float ww[V][B];           // dot(compressed, attn_w[O:])
  const int tid = threadIdx.x;
  const int wv = tid >> 5;
  const int lane = tid & 31;
  const int m = lane & 15;      // M row (A) / N col (B,C,D) within tile
  const int khalf = lane >> 4;  // K half select for A/B fragments
  const int v = wv;
  const float* sv = s + v * B * O;
  const float* w1v = se_w1 + v * H * O;   // [H][O]
  const float* w2v = se_w2 + v * O * H;   // [O][H]

  // --- matmul1: hpre[b][h] = sum_o s[b][o] * w1[h][o], M=32,N=16,K=64 ---
  for (int bt = 0; bt < 2; ++bt) {
    const int b0 = bt * 16;
    v8f acc = {};
    for (int kk = 0; kk < 16; ++kk) {
      const int k0 = kk * 4 + khalf * 2;
      v2f a, bfrag;
      a.x = sv[(b0 + m) * O + k0];        // A[m][k0]
      a.y = sv[(b0 + m) * O + k0 + 1];    // A[m][k0+1]
      bfrag.x = w1v[m * O + k0];          // B[k0][n=m] = w1[n][k0]
      bfrag.y = w1v[m * O + k0 + 1];
      acc = __builtin_amdgcn_wmma_f32_16x16x4_f32(false, a, false, bfrag,
                                                  (short)0, acc, false, false);
    }
#pragma unroll
    for (int r = 0; r < 8; ++r) {
      const int brow = b0 + r + khalf * 8;       // C/D: M = r (+8 hi half)
      const float hv = acc[r] + se_b1[v * H + m];
      hbuf[v][brow][m] = fmaxf(hv, 0.0f);
    }
  }
  __syncthreads();

  // --- matmul2: gpre[b][o] = sum_h h[b][h]*w2[o][h], M=32,N=64,K=16 ---
  for (int bt = 0; bt < 2; ++bt) {
    const int b0 = bt * 16;
    float ureg[8], wreg[8];
#pragma unroll
    for (int r = 0; r < 8; ++r) { ureg[r] = 0.f; wreg[r] = 0.f; }
    for (int ot = 0; ot < 4; ++ot) {
      const int o0 = ot * 16;
      v8f acc = {};
      for (int kk = 0; kk < 4; ++kk) {
        const int k0 = kk * 4 + khalf * 2;
        v2f a, bfrag;
        a.x = hbuf[v][b0 + m][k0];
        a.y = hbuf[v][b0 + m][k0 + 1];
        bfrag.x = w2v[(o0 + m) * H + k0];   // B[k0][n=m] = w2[o0+n][k0]
        bfrag.y = w2v[(o0 + m) * H + k0 + 1];
        acc = __builtin_amdgcn_wmma_f32_16x16x4_f32(false, a, false, bfrag,
                                                    (short)0, acc, false, false);
      }
#pragma unroll
      for (int r = 0; r < 8; ++r) {
        const int brow = b0 + r + khalf * 8;
        const int oc = o0 + m;
        const float gpre = acc[r] + se_b2[v * O + oc];
        const float gate = 1.0f / (1.0f + expf(-gpre));
        const float comp = gate * sv[brow * O + oc];  // compressed = gate*s
        ureg[r] += comp * attn_w[oc];
        wreg[r] += comp * attn_w[O + oc];
      }
    }
    // sum over the 16 lanes in each half (each half holds one set of b rows)
#pragma unroll
    for (int mask = 1; mask <= 8; mask <<= 1) {
#pragma unroll
      for (int r = 0; r < 8; ++r) {
        ureg[r] += __shfl_xor(ureg[r], mask);
        wreg[r] += __shfl_xor(wreg[r], mask);
      }
    }
    if (m == 0) {
#pragma unroll
      for (int r = 0; r < 8; ++r) {
        const int brow = b0 + r + khalf * 8;
        uu[v][brow] = ureg[r];
        ww[v][brow] = wreg[r];
      }
    }
  }
  __syncthreads();

  // --- softmax over v_from: thread (g = wv, b = lane) ---
  const int g = wv;
  const int b = lane;
  float awv[V];
  float mx = -3.4e38f;
#pragma unroll
  for (int f = 0; f < V; ++f) {
    const float a = selu_f(uu[f][b] + ww[g][b]);
    awv[f] = a;
    mx = fmaxf(mx, a);
  }
  float ssum = 0.f;
#pragma unroll
  for (int f = 0; f < V; ++f) { awv[f] = expf(awv[f] - mx); ssum += awv[f]; }
  const float inv = 1.0f / ssum;
#pragma unroll
  for (int f = 0; f < V; ++f) sm[(f * V + g) * B + b] = awv[f] * inv;
}

// ---------------------------------------------------------------------------
// Kernel C: grid (T/256, B). Recompute conv_f from xs on the fly and mix with
// sm; each thread owns one t, loops over o, writes 8 g-outputs (coalesced).
// ---------------------------------------------------------------------------
__global__ void mix_kernel(const float* __restrict__ xs,
                           const float* __restrict__ sm,
                           const float* __restrict__ conv_w,
                           const float* __restrict__ conv_b,
                           float* __restrict__ out) {
  __shared__ float xs_s[V][260];
  __shared__ float smv[V * V];
  const int b = blockIdx.y;
  const int t0 = blockIdx.x * 256;
  const int tid = threadIdx.x;
  if (tid < V * V) smv[tid] = sm[tid * B + b];  // smv[f*V+g]
  for (int i = tid; i < V * 260; i += 256) {
    const int f = i / 260;
    const int j = i % 260;
    const int t = t0 + j - 2;
    xs_s[f][j] = (t >= 0 && t < T) ? xs[(size_t)(f * B + b) * T + t] : 0.f;
  }
  __syncthreads();
  float xr[V][5];
#pragma unroll
  for (int f = 0; f < V; ++f)
#pragma unroll
    for (int k = 0; k < 5; ++k) xr[f][k] = xs_s[f][tid + k];
  const size_t tglob = (size_t)t0 + tid;
  for (int o = 0; o < O; ++o) {
    const float w0 = conv_w[o * 5 + 0], w1 = conv_w[o * 5 + 1],
                w2 = conv_w[o * 5 + 2], w3 = conv_w[o * 5 + 3],
                w4 = conv_w[o * 5 + 4];
    const float bo = conv_b[o];
    float accg[V];
#pragma unroll
    for (int g = 0; g < V; ++g) accg[g] = 0.f;
#pragma unroll
    for (int f = 0; f < V; ++f) {
      float cv = fmaf(xr[f][0], w0, fmaf(xr[f][1], w1, fmaf(xr[f][2], w2,
                 fmaf(xr[f][3], w3, fmaf(xr[f][4], w4, bo)))));
      cv = selu_f(cv);
#pragma unroll
      for (int g = 0; g < V; ++g) accg[g] = fmaf(smv[f * V + g], cv, accg[g]);
    }
#pragma unroll
    for (int g = 0; g < V; ++g)
      out[(size_t)((g * B + b) * O + o) * T + tglob] = accg[g];
  }
}

extern "C" void kernel_launch(void* const* d_in, const int* in_sizes, int n_in,
                              void* d_out, int out_size, void* d_ws,
                              size_t ws_size, hipStream_t stream) {
  (void)in_sizes; (void)n_in; (void)out_size; (void)ws_size;
  const float* x      = (const float*)d_in[0];
  const float* conv_w = (const float*)d_in[1];
  const float* conv_b = (const float*)d_in[2];
  const float* se_w1  = (const float*)d_in[3];
  const float* se_b1  = (const float*)d_in[4];
  const float* se_w2  = (const float*)d_in[5];
  const float* se_b2  = (const float*)d_in[6];
  const float* attn_w = (const float*)d_in[7];
  float* out = (float*)d_out;

  float* xs = (float*)d_ws;                       // V*B*T   = 4 MB
  float* s  = xs + (size_t)V * B * T;             // V*B*O   = 64 KB
  float* sm = s + (size_t)V * B * O;              // V*V*B   = 8 KB

  conv_stats_kernel<<<V * B, 256, 0, stream>>>(x, conv_w, conv_b, xs, s);
  se_attn_kernel<<<1, 256, 0, stream>>>(s, se_w1, se_b1, se_w2, se_b2,
                                        attn_w, sm);
  mix_kernel<<<dim3(T / 256, B), 256, 0, stream>>>(xs, sm, conv_w, conv_b,
                                                   out);
}